// ResidualVectorQuantizer_19653770346758
// MI455X (gfx1250) — compile-verified
//
#include <hip/hip_runtime.h>

typedef __attribute__((ext_vector_type(2))) float v2f;
typedef __attribute__((ext_vector_type(8))) float v8f;

#define NB   8
#define NT   4096
#define ND   128
#define NCB  8
#define NK   1024
#define NN   (NB * NT)            // 32768 rows

#define THREADS      256          // 8 wave32
#define ROWS_PER_WG  128          // one 16-row tile per wave
#define CHUNK        64           // codes staged per LDS buffer
#define NCHUNK       (NK / CHUNK) // 16
#define CTILES       (CHUNK / 16) // 4
#define F4PT         (CHUNK * ND / 4 / THREADS) // 8 float4 per thread per stage

__global__ __launch_bounds__(THREADS, 2)
void rvq_fused(const float* __restrict__ emb,
               const float* __restrict__ cbs,
               int*   __restrict__ codes_out,
               float* __restrict__ quant_out,
               float* __restrict__ loss_out)
{
    __shared__ float4 lds_cb4[2][CHUNK * ND / 4];  // 2 x 32KB codebook stage
    __shared__ float  lds_nsq[2][CHUNK];           // per-code squared norms
    __shared__ int    lds_idx[ROWS_PER_WG];        // per-row argmin winner
    __shared__ float  lds_red[THREADS / 32];       // loss reduction

    const int tid  = threadIdx.x;
    const int lane = tid & 31;
    const int wid  = tid >> 5;
    const int hi16 = lane >> 4;          // which 16-lane half
    const int lrow = lane & 15;          // fragment row / col within tile
    const int dsel = hi16 * 2;           // k-dims {0,1} vs {2,3} of each 4-step
    const int row  = blockIdx.x * ROWS_PER_WG + wid * 16 + lrow;

    // Residual lives in registers in WMMA A-fragment (16x4 f32) layout:
    // lane l holds dims 4*kk + dsel + {0,1} of row (l&15), for kk = 0..31.
    v2f res[32];
    {
        const float* ep = emb + (size_t)row * ND + dsel;
        #pragma unroll
        for (int kk = 0; kk < 32; ++kk)
            res[kk] = *(const v2f*)(ep + 4 * kk);
    }

    float  lacc = 0.0f;
    float4 pf[F4PT];
    int    buf = 0;

    for (int cb = 0; cb < NCB; ++cb) {
        const float4* gcb4 = (const float4*)cbs + (size_t)cb * NK * (ND / 4);

        // prefetch chunk 0 into registers
        #pragma unroll
        for (int i = 0; i < F4PT; ++i)
            pf[i] = gcb4[(size_t)tid * F4PT + i];

        float bestS[8];
        int   bestI[8];
        #pragma unroll
        for (int v = 0; v < 8; ++v) { bestS[v] = -3.4e38f; bestI[v] = 0; }

        for (int chunk = 0; chunk < NCHUNK; ++chunk) {
            // ---- stage registers -> LDS, compute -0.5*||c||^2 source data ----
            {
                float4* l4 = lds_cb4[buf];
                float p = 0.0f;
                #pragma unroll
                for (int i = 0; i < F4PT; ++i) {
                    float4 x = pf[i];
                    l4[tid * F4PT + i] = x;
                    p += x.x * x.x + x.y * x.y + x.z * x.z + x.w * x.w;
                }
                // 4 consecutive threads hold one code's 128 dims
                p += __shfl_xor(p, 1, 32);
                p += __shfl_xor(p, 2, 32);
                if ((tid & 3) == 0) lds_nsq[buf][tid >> 2] = p;
            }
            __syncthreads();   // staged chunk + norms visible (1 barrier/chunk)

            // issue next chunk's global loads; latency hidden by 128 WMMAs
            if (chunk + 1 < NCHUNK) {
                #pragma unroll
                for (int i = 0; i < F4PT; ++i)
                    pf[i] = gcb4[(size_t)(chunk + 1) * CHUNK * (ND / 4)
                                 + tid * F4PT + i];
            }

            // ---- score tiles: s = r . c - 0.5*||c||^2, argmax == argmin d2 ----
            const float* lc = (const float*)lds_cb4[buf];
            #pragma unroll
            for (int ct = 0; ct < CTILES; ++ct) {
                float nsq = lds_nsq[buf][ct * 16 + lrow];
                v8f acc;
                #pragma unroll
                for (int v = 0; v < 8; ++v) acc[v] = -0.5f * nsq;

                // B fragment (4x16): lane l reads dims 4*kk+dsel of code (ct*16 + l&15)
                const float* cl = lc + (ct * 16 + lrow) * ND + dsel;
                #pragma unroll
                for (int kk = 0; kk < 32; ++kk) {
                    v2f a = res[kk];
                    v2f b = *(const v2f*)(cl + 4 * kk);
                    acc = __builtin_amdgcn_wmma_f32_16x16x4_f32(
                        false, a, false, b, (short)0, acc, false, false);
                }

                const int code_id = chunk * CHUNK + ct * 16 + lrow;
                #pragma unroll
                for (int v = 0; v < 8; ++v) {
                    if (acc[v] > bestS[v]) { bestS[v] = acc[v]; bestI[v] = code_id; }
                }
            }
            buf ^= 1;
        }

        // ---- per-row argmax across 16 lane-slots (first-min tie-break) ----
        #pragma unroll
        for (int v = 0; v < 8; ++v) {
            float s = bestS[v];
            int   i = bestI[v];
            #pragma unroll
            for (int off = 1; off < 16; off <<= 1) {
                float s2 = __shfl_xor(s, off, 32);
                int   i2 = __shfl_xor(i, off, 32);
                if (s2 > s || (s2 == s && i2 < i)) { s = s2; i = i2; }
            }
            // lane 0 holds rows v (M=v), lane 16 holds rows v+8 (M=v+8)
            if (lrow == 0) lds_idx[wid * 16 + v + 8 * hi16] = i;
        }
        __syncthreads();

        const int code = lds_idx[wid * 16 + lrow];
        if (hi16 == 0) codes_out[(size_t)row * NCB + cb] = code;

        // ---- gather winner (L2-resident), residual -= q, loss += ||res||^2 ----
        const float* qp = cbs + ((size_t)cb * NK + code) * ND + dsel;
        #pragma unroll
        for (int kk = 0; kk < 32; ++kk) {
            v2f q = *(const v2f*)(qp + 4 * kk);
            res[kk].x -= q.x;
            res[kk].y -= q.y;
            lacc += res[kk].x * res[kk].x + res[kk].y * res[kk].y;
        }
        __syncthreads();
    }

    // ---- quantized = emb - residual_final (sum of all gathered codes) ----
    {
        const float* ep = emb + (size_t)row * ND + dsel;
        float*       qo = quant_out + (size_t)row * ND + dsel;
        #pragma unroll
        for (int kk = 0; kk < 32; ++kk) {
            v2f e = *(const v2f*)(ep + 4 * kk);
            v2f q;
            q.x = e.x - res[kk].x;
            q.y = e.y - res[kk].y;
            *(v2f*)(qo + 4 * kk) = q;
        }
    }

    // ---- commitment loss: block reduce, then global atomic add ----
    lacc += __shfl_xor(lacc, 1, 32);
    lacc += __shfl_xor(lacc, 2, 32);
    lacc += __shfl_xor(lacc, 4, 32);
    lacc += __shfl_xor(lacc, 8, 32);
    lacc += __shfl_xor(lacc, 16, 32);
    if (lane == 0) lds_red[wid] = lacc;
    __syncthreads();
    if (tid == 0) {
        float s = 0.0f;
        #pragma unroll
        for (int w = 0; w < THREADS / 32; ++w) s += lds_red[w];
        atomicAdd(loss_out, s * (1.0f / ((float)NN * (float)ND * (float)NCB)));
    }
}

extern "C" void kernel_launch(void* const* d_in, const int* in_sizes, int n_in,
                              void* d_out, int out_size, void* d_ws, size_t ws_size,
                              hipStream_t stream)
{
    const float* emb = (const float*)d_in[0];   // [8,4096,128] f32
    const float* cbs = (const float*)d_in[1];   // [8,1024,128] f32

    float* out   = (float*)d_out;
    int*   codes = (int*)out;                       // [N, NCB] int32 (bit-stored)
    float* quant = out + (size_t)NN * NCB;          // [N, D] f32
    float* loss  = quant + (size_t)NN * ND;         // scalar f32

    hipMemsetAsync(loss, 0, sizeof(float), stream); // graph-capturable; replay-safe
    rvq_fused<<<dim3(NN / ROWS_PER_WG), dim3(THREADS), 0, stream>>>(
        emb, cbs, codes, quant, loss);
}